// GIN_GatedAttn_51917564674533
// MI455X (gfx1250) — compile-verified
//
#include <hip/hip_runtime.h>
#include <cstddef>

#define N_NODES 20000
#define N_EDGES 160000
#define IN_DIM  2048
#define HID     512

typedef __attribute__((ext_vector_type(2))) float v2f;
typedef __attribute__((ext_vector_type(8))) float v8f;

// ---------------------------------------------------------------------------
// hardware f32 atomic add (global_atomic_add_f32, no CAS loop)
// ---------------------------------------------------------------------------
__device__ __forceinline__ void atomAddF(float* p, float v) {
    unsafeAtomicAdd(p, v);
}

// ---------------------------------------------------------------------------
// hin = x  (vectorized copy into workspace)
// ---------------------------------------------------------------------------
__global__ __launch_bounds__(256) void copy_x_kernel(const float* __restrict__ x,
                                                     float* __restrict__ hin,
                                                     long long n4) {
    long long i = (long long)blockIdx.x * blockDim.x + threadIdx.x;
    if (i < n4) {
        ((float4*)hin)[i] = ((const float4*)x)[i];
    }
}

// ---------------------------------------------------------------------------
// scatter-add: hin[dst] += x[src] ; one wave per edge, coalesced lanes
// ---------------------------------------------------------------------------
__global__ __launch_bounds__(256) void scatter_add_kernel(const float* __restrict__ x,
                                                          const int* __restrict__ ei,
                                                          float* __restrict__ hin) {
    int wave = threadIdx.x >> 5;
    int lane = threadIdx.x & 31;
    int e = blockIdx.x * 8 + wave;           // 8 waves per block
    if (e >= N_EDGES) return;
    int src = ei[e];
    int dst = ei[N_EDGES + e];
    const float* xs = x + (size_t)src * IN_DIM;
    float* hd = hin + (size_t)dst * IN_DIM;
    #pragma unroll 8
    for (int f = lane; f < IN_DIM; f += 32) {
        atomAddF(hd + f, xs[f]);
    }
}

// ---------------------------------------------------------------------------
// alpha[i] = ba
// ---------------------------------------------------------------------------
__global__ __launch_bounds__(256) void init_alpha_kernel(float* __restrict__ alpha,
                                                         const float* __restrict__ ba,
                                                         int n) {
    int i = blockIdx.x * blockDim.x + threadIdx.x;
    if (i < n) alpha[i] = ba[0];
}

// ---------------------------------------------------------------------------
// fp32 WMMA GEMM:  C[M,N] = epilogue(A[M,K] @ B[K,N] + bias)
//   block tile 64x128, BK=16, 8 waves as 2(M) x 4(N); each wave: 32x32 tile
//   (four 16x16 f32 accumulators, A/B fragments each reused twice).
//   LDS: ping-pong buffers; A row-major pad-20 (16B aligned rows, conflict-free
//   frag reads), B stored transposed pad-18 so every WMMA fragment is one
//   contiguous ds_load_b64. One barrier per K-chunk; next chunk's global loads
//   are issued into registers before the barrier (software pipeline).
// EPI 0: relu store   EPI 1: plain store
// EPI 2: alpha += sum_col tanh(v + bias[col]) * Wa[col]   (no C store)
// ---------------------------------------------------------------------------
template <int EPI>
__global__ __launch_bounds__(256) void gemm_wmma_kernel(const float* __restrict__ A,
                                                        const float* __restrict__ B,
                                                        const float* __restrict__ bias,
                                                        float* __restrict__ C,
                                                        const float* __restrict__ Wa,
                                                        float* __restrict__ alpha,
                                                        int M, int N, int K) {
    constexpr int BM = 64, BN = 128, BK = 16;
    __shared__ float As[2][BM][20];     // row-major, stride 20 (80B: 16B-aligned)
    __shared__ float BsT[2][BN][18];    // transposed, stride 18 (72B: 8B-aligned)

    const int tid   = threadIdx.x;
    const int lane  = tid & 31;
    const int wave  = tid >> 5;
    const int wm    = wave & 1;        // 0..1  (M sub-tile, 32 rows)
    const int wn    = wave >> 1;       // 0..3  (N sub-tile, 32 cols)
    const int l15   = lane & 15;
    const int halfk = lane >> 4;       // 0: lanes 0-15, 1: lanes 16-31

    const int bm = blockIdx.x * BM;
    const int bn = blockIdx.y * BN;

    v8f acc[2][2] = {};                // [mi][ni] : rows wm*32+mi*16, cols wn*32+ni*16

    // staging indices: A tile 64x16 = one float4/thread; B tile 16x128 = two
    const int ar   = tid >> 2;          // 0..63
    const int ac   = (tid & 3) * 4;     // 0,4,8,12
    const int bk0  = tid >> 5;          // 0..7
    const int bn40 = (tid & 31) * 4;    // 0..124
    const int bk1  = bk0 + 8;           // 8..15
    const int bn41 = bn40;

    const int arow = bm + ar;
    const bool aok = (arow < M);

    float4 av, bv0, bv1;
    // prologue: chunk 0 global -> regs
    av = make_float4(0.f, 0.f, 0.f, 0.f);
    if (aok) av = *(const float4*)(A + (size_t)arow * K + ac);
    bv0 = *(const float4*)(B + (size_t)bk0 * N + bn + bn40);
    bv1 = *(const float4*)(B + (size_t)bk1 * N + bn + bn41);

    const int nc = K / BK;
    for (int c = 0; c < nc; ++c) {
        const int p = c & 1;
        // stage regs -> LDS buffer p
        *(float4*)&As[p][ar][ac] = av;
        BsT[p][bn40 + 0][bk0] = bv0.x;
        BsT[p][bn40 + 1][bk0] = bv0.y;
        BsT[p][bn40 + 2][bk0] = bv0.z;
        BsT[p][bn40 + 3][bk0] = bv0.w;
        BsT[p][bn41 + 0][bk1] = bv1.x;
        BsT[p][bn41 + 1][bk1] = bv1.y;
        BsT[p][bn41 + 2][bk1] = bv1.z;
        BsT[p][bn41 + 3][bk1] = bv1.w;

        // prefetch chunk c+1 -> regs (overlaps with compute below)
        if (c + 1 < nc) {
            const int k0 = (c + 1) * BK;
            av = make_float4(0.f, 0.f, 0.f, 0.f);
            if (aok) av = *(const float4*)(A + (size_t)arow * K + k0 + ac);
            bv0 = *(const float4*)(B + (size_t)(k0 + bk0) * N + bn + bn40);
            bv1 = *(const float4*)(B + (size_t)(k0 + bk1) * N + bn + bn41);
        }
        __syncthreads();

        // compute on buffer p: 16 WMMAs, every fragment one contiguous b64 load
        const int ra = wm * 32 + l15;
        const int cb = wn * 32 + l15;
        #pragma unroll
        for (int kk = 0; kk < BK; kk += 4) {
            const int ka = kk + halfk * 2;          // always even
            v2f a0 = *(const v2f*)&As[p][ra][ka];
            v2f a1 = *(const v2f*)&As[p][ra + 16][ka];
            v2f b0 = *(const v2f*)&BsT[p][cb][ka];
            v2f b1 = *(const v2f*)&BsT[p][cb + 16][ka];
            acc[0][0] = __builtin_amdgcn_wmma_f32_16x16x4_f32(false, a0, false, b0,
                                                              (short)0, acc[0][0], false, false);
            acc[0][1] = __builtin_amdgcn_wmma_f32_16x16x4_f32(false, a0, false, b1,
                                                              (short)0, acc[0][1], false, false);
            acc[1][0] = __builtin_amdgcn_wmma_f32_16x16x4_f32(false, a1, false, b0,
                                                              (short)0, acc[1][0], false, false);
            acc[1][1] = __builtin_amdgcn_wmma_f32_16x16x4_f32(false, a1, false, b1,
                                                              (short)0, acc[1][1], false, false);
        }
    }

    // epilogue: C/D layout -> VGPR r holds row r (lanes 0-15) / row r+8 (lanes 16-31)
    const int rbase = bm + wm * 32 + halfk * 8;
    const int col0  = bn + wn * 32 + l15;

    if (EPI == 2) {
        const float bg0 = bias[col0];
        const float bg1 = bias[col0 + 16];
        const float wa0 = Wa[col0];
        const float wa1 = Wa[col0 + 16];
        #pragma unroll
        for (int mi = 0; mi < 2; ++mi) {
            #pragma unroll
            for (int r = 0; r < 8; ++r) {
                const int row = rbase + mi * 16 + r;
                float contrib = tanhf(acc[mi][0][r] + bg0) * wa0 +
                                tanhf(acc[mi][1][r] + bg1) * wa1;
                // reduce 16 columns (stays within each 16-lane half)
                contrib += __shfl_xor(contrib, 8, 32);
                contrib += __shfl_xor(contrib, 4, 32);
                contrib += __shfl_xor(contrib, 2, 32);
                contrib += __shfl_xor(contrib, 1, 32);
                if (l15 == 0 && row < M) atomAddF(alpha + row, contrib);
            }
        }
    } else {
        const float bia0 = bias[col0];
        const float bia1 = bias[col0 + 16];
        #pragma unroll
        for (int mi = 0; mi < 2; ++mi) {
            #pragma unroll
            for (int r = 0; r < 8; ++r) {
                const int row = rbase + mi * 16 + r;
                if (row < M) {
                    float v0 = acc[mi][0][r] + bia0;
                    float v1 = acc[mi][1][r] + bia1;
                    if (EPI == 0) { v0 = fmaxf(v0, 0.f); v1 = fmaxf(v1, 0.f); }
                    C[(size_t)row * N + col0]      = v0;
                    C[(size_t)row * N + col0 + 16] = v1;
                }
            }
        }
    }
}

// ---------------------------------------------------------------------------
// single-workgroup softmax over N_NODES scores
// ---------------------------------------------------------------------------
__global__ __launch_bounds__(1024) void softmax_kernel(const float* __restrict__ alpha,
                                                       float* __restrict__ att, int n) {
    __shared__ float red[1024];
    __shared__ float s_stat;
    const int tid = threadIdx.x;

    float m = -3.4e38f;
    for (int i = tid; i < n; i += 1024) m = fmaxf(m, alpha[i]);
    red[tid] = m;
    __syncthreads();
    for (int s = 512; s > 0; s >>= 1) {
        if (tid < s) red[tid] = fmaxf(red[tid], red[tid + s]);
        __syncthreads();
    }
    if (tid == 0) s_stat = red[0];
    __syncthreads();
    const float mx = s_stat;

    float sum = 0.f;
    for (int i = tid; i < n; i += 1024) sum += expf(alpha[i] - mx);
    red[tid] = sum;
    __syncthreads();
    for (int s = 512; s > 0; s >>= 1) {
        if (tid < s) red[tid] += red[tid + s];
        __syncthreads();
    }
    if (tid == 0) s_stat = red[0];
    __syncthreads();
    const float inv = 1.0f / s_stat;

    for (int i = tid; i < n; i += 1024) att[i] = expf(alpha[i] - mx) * inv;
}

// ---------------------------------------------------------------------------
static inline size_t alignUp(size_t v) { return (v + 255) & ~(size_t)255; }

extern "C" void kernel_launch(void* const* d_in, const int* in_sizes, int n_in,
                              void* d_out, int out_size, void* d_ws, size_t ws_size,
                              hipStream_t stream) {
    const float* x  = (const float*)d_in[0];
    const int*   ei = (const int*)d_in[1];
    const float* W1 = (const float*)d_in[2];
    const float* b1 = (const float*)d_in[3];
    const float* W2 = (const float*)d_in[4];
    const float* b2 = (const float*)d_in[5];
    const float* Wg = (const float*)d_in[6];
    const float* bg = (const float*)d_in[7];
    const float* Wa = (const float*)d_in[8];
    const float* ba = (const float*)d_in[9];

    float* h   = (float*)d_out;                      // [N_NODES, HID]
    float* att = h + (size_t)N_NODES * HID;          // [N_NODES]

    char* ws = (char*)d_ws;
    size_t off = 0;
    float* hin = (float*)(ws + off);                 // [N_NODES, IN_DIM]
    off += alignUp((size_t)N_NODES * IN_DIM * sizeof(float));
    float* h1 = (float*)(ws + off);                  // [N_NODES, HID]
    off += alignUp((size_t)N_NODES * HID * sizeof(float));
    float* alpha = (float*)(ws + off);               // [N_NODES]

    // 1) hin = x
    {
        long long n4 = (long long)N_NODES * IN_DIM / 4;
        int blocks = (int)((n4 + 255) / 256);
        copy_x_kernel<<<blocks, 256, 0, stream>>>(x, hin, n4);
    }
    // 2) hin[dst] += x[src] over edges (one wave per edge)
    scatter_add_kernel<<<N_EDGES / 8, 256, 0, stream>>>(x, ei, hin);

    dim3 grid1((N_NODES + 63) / 64, HID / 128);
    // 3) h1 = relu(hin @ W1 + b1)
    gemm_wmma_kernel<0><<<grid1, 256, 0, stream>>>(hin, W1, b1, h1, nullptr, nullptr,
                                                   N_NODES, HID, IN_DIM);
    // 4) h = h1 @ W2 + b2  (written straight into d_out)
    gemm_wmma_kernel<1><<<grid1, 256, 0, stream>>>(h1, W2, b2, h, nullptr, nullptr,
                                                   N_NODES, HID, HID);
    // 5) alpha = ba ; alpha[i] += sum_c tanh((h @ Wg + bg))[i,c] * Wa[c]
    init_alpha_kernel<<<(N_NODES + 255) / 256, 256, 0, stream>>>(alpha, ba, N_NODES);
    gemm_wmma_kernel<2><<<grid1, 256, 0, stream>>>(h, Wg, bg, nullptr, Wa, alpha,
                                                   N_NODES, HID, HID);
    // 6) att = softmax(alpha)
    softmax_kernel<<<1, 1024, 0, stream>>>(alpha, att, N_NODES);
}